// GraphSpectralFilterLayer_41077067219249
// MI455X (gfx1250) — compile-verified
//
#include <hip/hip_runtime.h>
#include <hip/hip_bf16.h>
#include <math.h>

// Problem constants (from reference): N=8192, C=4, F_IN=512, F_OUT=64, K=16
#define NV      8192      // n_vertices (also attention row length)
#define NC      4
#define FIN     512
#define FOUT    64
#define KTOP    16
#define NROWS   (NV * NC) // 32768 attention rows

typedef float v2f __attribute__((ext_vector_type(2)));
typedef float v8f __attribute__((ext_vector_type(8)));

// ---------------------------------------------------------------------------
// Kernel 1: h = x @ W   (8192x512 @ 512x64 fp32) via V_WMMA_F32_16X16X4_F32.
// One wave computes one 16x16 tile of h; K advances 4 per WMMA.
//   A 16x4 fp32 layout (ISA 7.12.2): lanes 0-15 -> M=lane, VGPR0/1 = K=0/1;
//                                    lanes 16-31 -> M=lane-16, VGPR0/1 = K=2/3.
//   B 4x16 fp32: row striped across lanes per VGPR; VGPR0 = K=(0|2), VGPR1 = K=(1|3).
//   C/D 16x16 fp32: VGPR v -> M=v (lanes 0-15) / M=v+8 (lanes 16-31), N=lane%16.
// ---------------------------------------------------------------------------
__global__ __launch_bounds__(256) void linear_wmma_kernel(
    const float* __restrict__ x,   // [NV, FIN]
    const float* __restrict__ W,   // [FIN, FOUT]
    float* __restrict__ h)         // [NV, FOUT]
{
    const int lane = threadIdx.x & 31;
    const int wave = threadIdx.x >> 5;
    const int tile = blockIdx.x * 8 + wave;     // 0 .. 2047
    const int m0   = (tile >> 2) * 16;          // 512 M-tiles
    const int n0   = (tile & 3) * 16;           // 4  N-tiles
    const int lm   = lane & 15;
    const int kh   = (lane >> 4) * 2;           // 0 or 2: which K-pair this half-wave holds

    v8f acc = {};
    const float* xrow = x + (size_t)(m0 + lm) * FIN;

    #pragma unroll 4
    for (int k = 0; k < FIN; k += 4) {
        const int kk = k + kh;                  // even -> 8B aligned
        const float2 av = *(const float2*)(xrow + kk);
        v2f a; a.x = av.x; a.y = av.y;          // A[m][kk], A[m][kk+1]
        v2f b;
        b.x = W[(size_t)kk * FOUT + n0 + lm];   // B[kk][n]
        b.y = W[(size_t)(kk + 1) * FOUT + n0 + lm];
        acc = __builtin_amdgcn_wmma_f32_16x16x4_f32(
            /*neg_a=*/false, a, /*neg_b=*/false, b,
            /*c_mod=*/(short)0, acc, /*reuse_a=*/false, /*reuse_b=*/false);
    }

    #pragma unroll
    for (int v = 0; v < 8; ++v) {
        const int mrow = m0 + v + ((lane >> 4) << 3);
        h[(size_t)mrow * FOUT + n0 + lm] = acc[v];
    }
}

// ---------------------------------------------------------------------------
// Kernel 2: fused top-16 + softmax + spmm + permuted store.
// One wave per attention row (32768 rows). Single streaming pass over the
// 1.07 GB attention matrix (the bandwidth-bound term: ~46 us floor @ 23.3 TB/s).
// ---------------------------------------------------------------------------
__global__ __launch_bounds__(256) void topk_spmm_kernel(
    const float* __restrict__ attn,  // [NROWS, NV]
    const float* __restrict__ h,     // [NV, FOUT]
    float* __restrict__ out)         // [NV, NC*FOUT]
{
    const int   lane = threadIdx.x & 31;
    const int   wave = threadIdx.x >> 5;
    const int   row  = blockIdx.x * 8 + wave;          // 0 .. 32767
    const float* arow = attn + (size_t)row * NV;

    const float NEG = -__builtin_inff();

    // ---- per-lane register top-16 over this lane's 256 columns ----
    float tv[KTOP];
    int   ti[KTOP];
    #pragma unroll
    for (int k = 0; k < KTOP; ++k) { tv[k] = NEG; ti[k] = 0; }
    float minv = NEG;
    int   minp = 0;

    // lane owns columns [t*128 + lane*4, +4): fully coalesced b128 loads
    for (int t = 0; t < NV / 128; ++t) {
        const float4 vv = ((const float4*)arow)[t * 32 + lane];
        __builtin_prefetch(arow + (t + 8) * 128 + lane * 4, 0, 1);  // global_prefetch_b8
        const int jb = t * 128 + lane * 4;
        float e4[4] = { vv.x, vv.y, vv.z, vv.w };
        #pragma unroll
        for (int q = 0; q < 4; ++q) {
            const float v = e4[q];
            if (v > minv) {                      // rare after warm-up
                #pragma unroll
                for (int k = 0; k < KTOP; ++k)
                    if (k == minp) { tv[k] = v; ti[k] = jb + q; }
                minv = tv[0]; minp = 0;
                #pragma unroll
                for (int k = 1; k < KTOP; ++k)
                    if (tv[k] < minv) { minv = tv[k]; minp = k; }
            }
        }
    }

    // ---- 16 rounds of wave-wide argmax (descending) fused with softmax+spmm ----
    float2 acc = make_float2(0.f, 0.f);
    float  m = 0.f, sum = 0.f;

    #pragma unroll
    for (int r = 0; r < KTOP; ++r) {
        // lane-local argmax over remaining candidates (register selects)
        float bv = NEG; int bp = 0;
        #pragma unroll
        for (int k = 0; k < KTOP; ++k)
            if (tv[k] > bv) { bv = tv[k]; bp = k; }
        int bi = 0;
        #pragma unroll
        for (int k = 0; k < KTOP; ++k)
            if (k == bp) bi = ti[k];

        // wave32 argmax reduction, tie-break on smaller index (top_k semantics)
        float v = bv; int i = bi;
        #pragma unroll
        for (int off = 16; off > 0; off >>= 1) {
            const float ov = __shfl_xor(v, off, 32);
            const int   oi = __shfl_xor(i, off, 32);
            if (ov > v || (ov == v && (unsigned)oi < (unsigned)i)) { v = ov; i = oi; }
        }

        // winner retires its candidate
        if (bv == v && bi == i) {
            #pragma unroll
            for (int k = 0; k < KTOP; ++k)
                if (k == bp) tv[k] = NEG;
        }

        // fused softmax numerator + gather-accumulate (i is wave-uniform)
        if (r == 0) m = v;
        const float e = __expf(v - m);
        sum += e;
        const float2 hv = ((const float2*)(h + (size_t)i * FOUT))[lane];
        acc.x += e * hv.x;
        acc.y += e * hv.y;
    }

    const float inv = 1.0f / sum;
    acc.x *= inv;
    acc.y *= inv;

    // out[n, c*FOUT + f] with row = c*NV + n ; lane owns f = 2*lane, 2*lane+1
    const int n = row & (NV - 1);
    const int c = row >> 13;
    float2* orow = (float2*)(out + (size_t)n * (NC * FOUT) + c * FOUT);
    orow[lane] = acc;
}

extern "C" void kernel_launch(void* const* d_in, const int* in_sizes, int n_in,
                              void* d_out, int out_size, void* d_ws, size_t ws_size,
                              hipStream_t stream) {
    const float* x    = (const float*)d_in[0];   // [8192, 512]
    const float* W    = (const float*)d_in[1];   // [512, 64]
    const float* attn = (const float*)d_in[2];   // [32768, 8192]
    float*       out  = (float*)d_out;           // [8192, 256]
    float*       h    = (float*)d_ws;            // [8192, 64] fp32 = 2 MB scratch

    // h = x @ W  (WMMA fp32): 2048 tile-waves -> 256 blocks x 8 waves
    linear_wmma_kernel<<<256, 256, 0, stream>>>(x, W, h);

    // fused top-k + softmax + spmm: 32768 rows, 1 wave/row, 8 waves/block
    topk_spmm_kernel<<<4096, 256, 0, stream>>>(attn, h, out);
}